// SimpleScanModel_36395552867092
// MI455X (gfx1250) — compile-verified
//
#include <hip/hip_runtime.h>

typedef __attribute__((ext_vector_type(2))) float v2f;
typedef __attribute__((ext_vector_type(8))) float v8f;
typedef int v4i __attribute__((vector_size(16)));   // matches builtin param type

#define TILE 8192   // elements per block tile (256 threads * 32)

#define AS1 __attribute__((address_space(1)))
#define AS3 __attribute__((address_space(3)))

#if __has_builtin(__builtin_amdgcn_global_load_async_to_lds_b128)
#define HAVE_ASYNC_LDS 1
#else
#define HAVE_ASYNC_LDS 0
#warning "no __builtin_amdgcn_global_load_async_to_lds_b128 on this toolchain; using sync staging"
#endif

static __device__ __forceinline__ void wait_asynccnt0() {
#if __has_builtin(__builtin_amdgcn_s_wait_asynccnt)
  __builtin_amdgcn_s_wait_asynccnt(0);
#else
  asm volatile("s_wait_asynccnt 0x0" ::: "memory");
#endif
}

// ---------------- Kernel 1: per-tile sums (HBM read pass) ----------------
__global__ __launch_bounds__(256) void k1_tile_sums(const float* __restrict__ x,
                                                    float* __restrict__ bs, int n) {
  int t = threadIdx.x;
  int b = blockIdx.x;
  long base = (long)b * TILE;
  __shared__ float sh[256];

  // warm L2 for the next tile (global_prefetch_b8 path)
  if (base + TILE < n && t == 0) __builtin_prefetch((const void*)(x + base + TILE), 0, 1);

  const float4* x4 = (const float4*)x;
  long base4 = base >> 2;
  int n4 = n >> 2;
  float s = 0.f;
#pragma unroll
  for (int i = 0; i < 8; i++) {
    long idx = base4 + t + i * 256;
    if (idx < n4) {
      float4 v = x4[idx];
      s += v.x; s += v.y; s += v.z; s += v.w;
    }
  }
  sh[t] = s;
  __syncthreads();
  for (int d = 128; d > 0; d >>= 1) {
    if (t < d) sh[t] += sh[t + d];
    __syncthreads();
  }
  if (t == 0) bs[b] = sh[0];
}

// ---------------- Kernel 2: scan of tile sums (single block) ----------------
__global__ __launch_bounds__(256) void k2_scan_sums(const float* __restrict__ bs,
                                                    float* __restrict__ boff, int nb) {
  __shared__ float ts[256];
  int t = threadIdx.x;
  float v[16];
  float run = 0.f;
#pragma unroll
  for (int i = 0; i < 16; i++) {
    int idx = t * 16 + i;
    v[i] = (idx < nb) ? bs[idx] : 0.f;
    run += v[i];
  }
  ts[t] = run;
  __syncthreads();
  // Hillis-Steele over 256 thread totals (deterministic)
  for (int d = 1; d < 256; d <<= 1) {
    float add = (t >= d) ? ts[t - d] : 0.f;
    __syncthreads();
    ts[t] += add;
    __syncthreads();
  }
  float excl = (t > 0) ? ts[t - 1] : 0.f;
  float run2 = excl;
#pragma unroll
  for (int i = 0; i < 16; i++) {
    int idx = t * 16 + i;
    if (idx < nb) boff[idx] = run2;  // exclusive tile offsets
    run2 += v[i];
  }
}

// ---------------- Kernel 3: WMMA intra-tile scan + offset (L2 read + HBM write) ----
__global__ __launch_bounds__(256) void k3_scan_tiles(const float* __restrict__ x,
                                                     const float* __restrict__ boff,
                                                     float* __restrict__ out, int n) {
  __shared__ float tile[TILE];     // 32 KB staged tile
  __shared__ float pw[256];        // per-thread partials
  __shared__ float woff[8];        // per-wave exclusive offsets
  __shared__ float rsArr[8 * 16];  // per-wave row-sum scratch (C->B relayout)

  int t = threadIdx.x;
  int b = blockIdx.x;
  long base = (long)b * TILE;

  // ---- stage tile into LDS ----
  {
    long base4 = base >> 2;   // float4 index of tile start
    int n4 = n >> 2;
#if HAVE_ASYNC_LDS
    bool full = ((base + TILE) <= (long)n);  // uniform across block
    if (full) {
      // CDNA5 async DMA path: global -> LDS, no VGPR round trip (ASYNCcnt)
#pragma unroll
      for (int i = 0; i < 8; i++) {
        long fidx = (base4 + t + i * 256) << 2;          // float index, 16B aligned
        AS1 v4i* g = (AS1 v4i*)(x + fidx);
        AS3 v4i* l = (AS3 v4i*)(tile + ((t + i * 256) << 2));
        __builtin_amdgcn_global_load_async_to_lds_b128(g, l, 0, 0);
      }
      wait_asynccnt0();
    } else
#endif
    {
      const float4* x4 = (const float4*)x;
      float4* t4 = (float4*)tile;
#pragma unroll
      for (int i = 0; i < 8; i++) {
        long idx = base4 + t + i * 256;
        float4 v;
        if (idx < n4) v = x4[idx];
        else { v.x = 0.f; v.y = 0.f; v.z = 0.f; v.w = 0.f; }
        t4[t + i * 256] = v;
      }
    }
  }
  __syncthreads();

  // ---- per-wave totals (fixed order) -> wave offsets ----
  {
    float s = 0.f;
    int o = t * 32;
#pragma unroll
    for (int j = 0; j < 32; j++) s += tile[o + j];
    pw[t] = s;
  }
  __syncthreads();
  if (t == 0) {
    float run = 0.f;
    for (int w = 0; w < 8; w++) {
      float ws = 0.f;
      for (int j = 0; j < 32; j++) ws += pw[w * 32 + j];
      woff[w] = run;
      run += ws;
    }
  }
  __syncthreads();

  int w = t >> 5, lane = t & 31, half = lane >> 4, col = lane & 15;
  float runOff = boff[b] + woff[w];

  // Constant matrices in native WMMA layouts (wave32):
  //   A-layout 16x4 chunk k: lanes 0-15 -> K=4k,4k+1 ; lanes 16-31 -> K=4k+2,4k+3 ; M = lane&15
  //   B-layout 4x16 chunk k: same K split across lane halves ; N = lane&15
  v2f bU[4], aL[4];
#pragma unroll
  for (int k = 0; k < 4; k++) {
    int r0 = 4 * k + 2 * half;
    bU[k].x = (col >= r0)     ? 1.f : 0.f;  // U incl. diagonal (row-wise inclusive prefix)
    bU[k].y = (col >= r0 + 1) ? 1.f : 0.f;
    aL[k].x = (r0     < col)  ? 1.f : 0.f;  // strict lower triangular ones
    aL[k].y = (r0 + 1 < col)  ? 1.f : 0.f;
  }

  // ---- each wave: 4 chained 256-element WMMA segment scans ----
#pragma unroll
  for (int s = 0; s < 4; s++) {
    int segBase = w * 1024 + s * 256;

    // X in A-layout: each lane reads 4 aligned float2 pairs from LDS
    v2f aX[4];
#pragma unroll
    for (int k = 0; k < 4; k++)
      aX[k] = *(const v2f*)&tile[segBase + col * 16 + 4 * k + 2 * half];

    // C = X * U  (row-wise inclusive prefixes; col 15 = row sums)
    v8f c = {};
#pragma unroll
    for (int k = 0; k < 4; k++)
      c = __builtin_amdgcn_wmma_f32_16x16x4_f32(false, aX[k], false, bU[k],
                                                (short)0, c, false, false);

    // extract row sums (C-layout col 15 lives in lanes 15 and 31)
    if (col == 15) {
#pragma unroll
      for (int r = 0; r < 8; r++) rsArr[w * 16 + half * 8 + r] = c[r];
    }
    // rebuild row sums as a broadcast matrix in B-layout
    v2f bM[4];
#pragma unroll
    for (int k = 0; k < 4; k++) {
      int r0 = 4 * k + 2 * half;
      bM[k].x = rsArr[w * 16 + r0];
      bM[k].y = rsArr[w * 16 + r0 + 1];
    }

    // C += Lstrict * R  (adds preceding-row offsets)
#pragma unroll
    for (int k = 0; k < 4; k++)
      c = __builtin_amdgcn_wmma_f32_16x16x4_f32(false, aL[k], false, bM[k],
                                                (short)0, c, false, false);

    // add running offset and store (half-wave contiguous b32 stores)
#pragma unroll
    for (int r = 0; r < 8; r++) c[r] += runOff;
    long outBase = base + segBase;
#pragma unroll
    for (int r = 0; r < 8; r++) {
      long idx = outBase + (long)(r + 8 * half) * 16 + col;
      if (idx < n) out[idx] = c[r];
    }

    // next segment offset = inclusive value at element 255 (row 15, col 15 = lane 31, VGPR 7)
    runOff = __int_as_float(__builtin_amdgcn_readlane(__float_as_int(c[7]), 31));
  }
}

// ---------------- launcher ----------------
extern "C" void kernel_launch(void* const* d_in, const int* in_sizes, int n_in,
                              void* d_out, int out_size, void* d_ws, size_t ws_size,
                              hipStream_t stream) {
  const float* x = (const float*)d_in[0];
  float* out = (float*)d_out;
  int n = in_sizes[0];
  int nb = (n + TILE - 1) / TILE;  // 4096 for N = 2^25

  float* bs = (float*)d_ws;    // tile sums
  float* boff = bs + nb;       // exclusive tile offsets

  k1_tile_sums<<<nb, 256, 0, stream>>>(x, bs, n);
  k2_scan_sums<<<1, 256, 0, stream>>>(bs, boff, nb);
  k3_scan_tiles<<<nb, 256, 0, stream>>>(x, boff, out, n);
}